// WeightedChamferLoss_59923383714444
// MI455X (gfx1250) — compile-verified
//
#include <hip/hip_runtime.h>
#include <math.h>

// ---------------------------------------------------------------------------
// Chamfer(k=16), B=2, N=M=8192, both directions.
//
// Distance tiles via V_WMMA_F32_16X16X4_F32:
//   A[m] = (rx, ry, rz, |r|^2)     (16 refs x K=4, staged in LDS)
//   B[n] = (-2qx,-2qy,-2qz, 1)     (K=4 x 16 queries, registers)
//   C[m][n] = |q_n|^2              (per-lane broadcast -> D = true d^2 >= 0)
//
// Top-16 per lane kept as ascending *unsigned bit patterns* (d^2 >= 0 is
// monotone under u32 compare) -> v_min_u32/v_max_u32 insertion network with
// no fp canonicalization, execz-guarded. Tiles processed in pairs so insert
// VALU of tile t overlaps the WMMA of tile t+1. Final half-merge between
// lanes l and l+16: union's 16 smallest = min(a[i], b[15-i]).
// ---------------------------------------------------------------------------

typedef float v2f __attribute__((ext_vector_type(2)));
typedef float v8f __attribute__((ext_vector_type(8)));

#define BATCH   2
#define NPTS    8192
#define MPTS    8192
#define KNN     16
#define CHUNK   2048          // refs staged per LDS pass (32 KB as float4)
#define TILES   (CHUNK / 16)  // 128 WMMA tiles per chunk
#define WAVES   4
#define BLOCK   (WAVES * 32)  // 128 threads
#define QPB     (WAVES * 16)  // 64 queries per workgroup

__device__ __forceinline__ void insert16u(unsigned (&top)[16], unsigned v) {
    // top[] ascending unsigned; keep 16 smallest. Whole-wave skip via execz.
    if (v < top[15]) {
        unsigned x = v;
#pragma unroll
        for (int i = 0; i < 16; ++i) {
            unsigned mn = (x < top[i]) ? x : top[i];   // v_min_u32
            unsigned mx = (x < top[i]) ? top[i] : x;   // v_max_u32
            top[i] = mn;
            x = mx;
        }
    }
}

__global__ void __launch_bounds__(BLOCK)
chamfer_knn_kernel(const float* __restrict__ pred,   // B*N*3 (source+flow)
                   const float* __restrict__ tgt,    // B*M*3
                   float* __restrict__ accum)        // scalar accumulator
{
    __shared__ float4 sref[CHUNK];                   // (x,y,z,|r|^2)

    const int lane = threadIdx.x & 31;
    const int wave = threadIdx.x >> 5;
    const int b    = blockIdx.y;
    const int dir  = blockIdx.z;

    const float* query = (dir == 0) ? pred : tgt;
    const float* ref   = (dir == 0) ? tgt  : pred;
    const float* qb = query + (size_t)b * NPTS * 3;
    const float* rb = ref   + (size_t)b * MPTS * 3;

    // ---- per-lane query (lanes l and l+16 share query l) ----
    const int qi = blockIdx.x * QPB + wave * 16 + (lane & 15);
    const float qx = qb[3 * qi + 0];
    const float qy = qb[3 * qi + 1];
    const float qz = qb[3 * qi + 2];
    const float q2 = qx * qx + qy * qy + qz * qz;

    // B operand: lanes 0-15 -> (K0,K1)=(-2qx,-2qy); lanes 16-31 -> (K2,K3)=(-2qz,1)
    v2f bop;
    if (lane < 16) { bop[0] = -2.0f * qx; bop[1] = -2.0f * qy; }
    else           { bop[0] = -2.0f * qz; bop[1] = 1.0f;       }

    // C operand: |q|^2 broadcast -> WMMA emits true squared distances
    v8f cq;
#pragma unroll
    for (int e = 0; e < 8; ++e) cq[e] = q2;

    unsigned top[16];
#pragma unroll
    for (int i = 0; i < 16; ++i) top[i] = 0x7F800000u;  // +inf bit pattern

    const float* sf = reinterpret_cast<const float*>(sref);
    const int alo = ((lane & 15) << 2) + ((lane >> 4) << 1); // lane's A offset

    for (int c0 = 0; c0 < MPTS; c0 += CHUNK) {
        __syncthreads();  // previous chunk's LDS reads done
        for (int r = threadIdx.x; r < CHUNK; r += BLOCK) {
            const float* rp = rb + (size_t)(c0 + r) * 3;
            float x = rp[0], y = rp[1], z = rp[2];
            sref[r] = make_float4(x, y, z, x * x + y * y + z * z);
        }
        __syncthreads();

        // tiles in pairs: both ds_loads + both WMMAs issue first, then the
        // insert VALU of tile t runs while tile t+1's WMMA is in flight.
        for (int t = 0; t < TILES; t += 2) {
            const int o0 = (t << 6) + alo;        // t*16*4 floats
            const int o1 = o0 + 64;
            v2f a0, a1;
            a0[0] = sf[o0];     a0[1] = sf[o0 + 1];
            a1[0] = sf[o1];     a1[1] = sf[o1 + 1];

            v8f d0 = __builtin_amdgcn_wmma_f32_16x16x4_f32(
                false, a0, false, bop, (short)0, cq, false, false);
            v8f d1 = __builtin_amdgcn_wmma_f32_16x16x4_f32(
                false, a1, false, bop, (short)0, cq, false, false);

#pragma unroll
            for (int e = 0; e < 8; ++e) insert16u(top, __float_as_uint(d0[e]));
#pragma unroll
            for (int e = 0; e < 8; ++e) insert16u(top, __float_as_uint(d1[e]));
        }
    }

    // ---- merge the two half-lists (lanes l <-> l+16): union's 16 smallest ----
    float s = 0.0f;
#pragma unroll
    for (int i = 0; i < 16; ++i) {
        unsigned o = __shfl_xor(top[15 - i], 16);
        unsigned u = (o < top[i]) ? o : top[i];
        s += sqrtf(fmaxf(__uint_as_float(u), 0.0f));
    }
    s *= (1.0f / (float)KNN);          // mean over k -> per-query value
    if (lane >= 16) s = 0.0f;          // lanes l / l+16 computed identical sums

    // wave32 reduction, one atomic per wave
#pragma unroll
    for (int off = 16; off >= 1; off >>= 1) s += __shfl_xor(s, off);
    if (lane == 0) atomicAdd(accum, s);
}

__global__ void prep_kernel(const float* __restrict__ src,
                            const float* __restrict__ flow,
                            float* __restrict__ pred,
                            float* __restrict__ accum, int n)
{
    int i = blockIdx.x * blockDim.x + threadIdx.x;
    if (i == 0) *accum = 0.0f;                 // re-init every call (determinism)
    if (i < n) pred[i] = src[i] + flow[i];
}

__global__ void finalize_kernel(const float* __restrict__ accum,
                                float* __restrict__ out)
{
    out[0] = accum[0] * (1.0f / (float)(BATCH * NPTS));
}

extern "C" void kernel_launch(void* const* d_in, const int* in_sizes, int n_in,
                              void* d_out, int out_size, void* d_ws, size_t ws_size,
                              hipStream_t stream)
{
    const float* src  = (const float*)d_in[0];   // pc_source [B,N,3]
    const float* tgt  = (const float*)d_in[1];   // pc_target [B,M,3]
    const float* flow = (const float*)d_in[2];   // pred_flow [B,N,3]
    float* out = (float*)d_out;

    char*  ws    = (char*)d_ws;
    float* accum = (float*)ws;                   // 1 float
    float* pred  = (float*)(ws + 256);           // B*N*3 floats

    const int n = BATCH * NPTS * 3;
    prep_kernel<<<(n + 255) / 256, 256, 0, stream>>>(src, flow, pred, accum, n);

    dim3 grid(NPTS / QPB, BATCH, 2);             // 128 x 2 x 2 = 512 WGs
    chamfer_knn_kernel<<<grid, BLOCK, 0, stream>>>(pred, tgt, accum);

    finalize_kernel<<<1, 1, 0, stream>>>(accum, out);
}